// deform_conv_v1_23063974380202
// MI455X (gfx1250) — compile-verified
//
#include <hip/hip_runtime.h>

// Deformable conv fused kernel for MI455X (gfx1250, wave32, WMMA).
// Shapes: x(8,64,128,128) w_off(18,64,3,3) b_off(18) w_conv(64,64,3,3) b_conv(64)
// out(8,64,128,128) fp32.

#define C_IN      64
#define C_OUT     64
#define HW        128
#define IMG       (HW * HW)      // 16384
#define PH        130            // padded height/width
#define NPIX_BLK  64             // pixels per workgroup
#define THREADS   128            // 4 waves of 32
#define CI_CHUNK  8
#define KC        (CI_CHUNK * 9) // 72, K-chunk of the implicit GEMM
#define A_STRIDE  73             // 64 x 73 patch tile (bank-conflict-free frag loads)
#define B_PS      160            // floats per K-PAIR row of B (PS%64==32 -> half-waves disjoint)
#define NSAMP     (NPIX_BLK * 9) // 576 bilinear sample points per block

typedef float v2f __attribute__((ext_vector_type(2)));
typedef float v8f __attribute__((ext_vector_type(8)));

__global__ __launch_bounds__(THREADS, 2)
void deform_conv_fused(const float* __restrict__ x,
                       const float* __restrict__ w_off,
                       const float* __restrict__ b_off,
                       const float* __restrict__ w_conv,
                       const float* __restrict__ b_conv,
                       float* __restrict__ out)
{
    // A tile (64x73 f32, 18688B). Reused at the end as output staging (64x65).
    __shared__ float sA[NPIX_BLK * A_STRIDE];
    // B tile, K-pair interleaved: sB[(k>>1)*B_PS + 2*co + (k&1)], 36x160 f32 = 23040B.
    // Reused in phase 1/2 as raw offsets (64x18).
    __shared__ float sB[(KC / 2) * B_PS];
    // Per-sample bilinear metadata.
    __shared__ int   sX0[NSAMP];
    __shared__ int   sY0[NSAMP];
    __shared__ float sWt[NSAMP * 4];

    const int tid  = threadIdx.x;
    const int lane = tid & 31;
    const int wv   = tid >> 5;        // wave id 0..3 -> M-tile
    const int half = lane >> 4;       // 0: K=k0,k0+1  1: K=k0+2,k0+3
    const int row  = lane & 15;       // M index (A) and N index (B/C/D)

    const int pix0 = blockIdx.x * NPIX_BLK;   // flat pixel = n*16384 + h*128 + w
    const int n    = pix0 >> 14;              // 16384 % 64 == 0 -> uniform per block
    const int pin0 = pix0 & (IMG - 1);

    const float* __restrict__ xn = x + (size_t)n * C_IN * IMG;

    // ---------------- Phase 1: offset conv (18 ch) for this block's 64 pixels ----
    {
        float* sOff = sB;                       // 64*18 floats alias into sB
        const int pl  = tid >> 1;               // pixel 0..63
        const int chb = (tid & 1) * 9;          // channels [chb, chb+9)
        const int pg  = pix0 + pl;
        const int h   = (pg & (IMG - 1)) >> 7;
        const int w   = pg & (HW - 1);
        float acc9[9];
        #pragma unroll
        for (int c = 0; c < 9; ++c) acc9[c] = b_off[chb + c];
        for (int ci = 0; ci < C_IN; ++ci) {
            const float* xc = xn + ci * IMG;
            float xv[9];
            #pragma unroll
            for (int a = 0; a < 3; ++a) {
                #pragma unroll
                for (int b = 0; b < 3; ++b) {
                    const int yy = h + a - 1, xx = w + b - 1;
                    const bool ok = ((unsigned)yy < (unsigned)HW) &
                                    ((unsigned)xx < (unsigned)HW);
                    xv[a * 3 + b] = ok ? xc[yy * HW + xx] : 0.0f;
                }
            }
            #pragma unroll
            for (int c = 0; c < 9; ++c) {
                const float* wrow = w_off + ((size_t)(chb + c) * C_IN + ci) * 9;
                #pragma unroll
                for (int ab = 0; ab < 9; ++ab) acc9[c] = fmaf(wrow[ab], xv[ab], acc9[c]);
            }
        }
        #pragma unroll
        for (int c = 0; c < 9; ++c) sOff[pl * 18 + chb + c] = acc9[c];
    }
    __syncthreads();

    // ---------------- Phase 2: sampling coords / bilinear weights ----------------
    {
        const float* sOff = sB;
        for (int s = tid; s < NSAMP; s += THREADS) {
            const int p = s / 9, kidx = s - p * 9;
            const int ki = kidx / 3, kj = kidx - ki * 3;
            const int pg = pix0 + p;
            const int h  = (pg & (IMG - 1)) >> 7;
            const int w  = pg & (HW - 1);
            // PAD + w + (kj - 1) == w + kj ; same for cy.
            float cx = sOff[p * 18 + kidx]     + (float)(w + kj);
            float cy = sOff[p * 18 + 9 + kidx] + (float)(h + ki);
            cx = fminf(fmaxf(cx, 0.0f), (float)(PH - 1));
            cy = fminf(fmaxf(cy, 0.0f), (float)(PH - 1));
            const float fx0 = floorf(cx), fy0 = floorf(cy);
            const float wx = cx - fx0, wy = cy - fy0;
            sX0[s] = (int)fx0;
            sY0[s] = (int)fy0;
            sWt[s * 4 + 0] = (1.0f - wx) * (1.0f - wy);
            sWt[s * 4 + 1] = wx * (1.0f - wy);
            sWt[s * 4 + 2] = (1.0f - wx) * wy;
            sWt[s * 4 + 3] = wx * wy;
        }
    }
    __syncthreads();

    // ---------------- Phase 3: implicit GEMM with fp32 WMMA ----------------------
    // out[p][co] = sum_k A[p][k] * B[k][co],  k = ci*9 + tap
    v8f acc[4];
    {
        const v8f vz = {0.f, 0.f, 0.f, 0.f, 0.f, 0.f, 0.f, 0.f};
        #pragma unroll
        for (int t = 0; t < 4; ++t) acc[t] = vz;
    }
    const float* Arow = &sA[(wv * 16 + row) * A_STRIDE];

    for (int ci0 = 0; ci0 < C_IN; ci0 += CI_CHUNK) {
        // Fill B tile, K-pair interleaved (co varies across lanes -> distinct banks).
        for (int idx = tid; idx < KC * C_OUT; idx += THREADS) {
            const int co = idx & (C_OUT - 1);
            const int j  = idx >> 6;                         // k within chunk, 0..71
            sB[(j >> 1) * B_PS + (co << 1) + (j & 1)] =
                w_conv[(size_t)co * (C_IN * 9) + ci0 * 9 + j];
        }
        // Fill A tile: bilinear sample 8 channels per (pixel, tap).
        for (int s = tid; s < NSAMP; s += THREADS) {
            const int p = s / 9, kidx = s - p * 9;
            const int x0 = sX0[s], y0 = sY0[s];
            const int x1 = min(x0 + 1, PH - 1), y1 = min(y0 + 1, PH - 1);
            const float w00 = sWt[s * 4 + 0], w01 = sWt[s * 4 + 1];
            const float w10 = sWt[s * 4 + 2], w11 = sWt[s * 4 + 3];
            // padded coords -> unpadded x coords (zero outside)
            const int xx0 = x0 - 1, yy0 = y0 - 1, xx1 = x1 - 1, yy1 = y1 - 1;
            const bool vx0 = (unsigned)xx0 < (unsigned)HW;
            const bool vx1 = (unsigned)xx1 < (unsigned)HW;
            const bool vy0 = (unsigned)yy0 < (unsigned)HW;
            const bool vy1 = (unsigned)yy1 < (unsigned)HW;
            const bool v00 = vy0 & vx0, v01 = vy0 & vx1;
            const bool v10 = vy1 & vx0, v11 = vy1 & vx1;
            const int o00 = yy0 * HW + xx0, o01 = yy0 * HW + xx1;
            const int o10 = yy1 * HW + xx0, o11 = yy1 * HW + xx1;
            const float* xc = xn + (size_t)ci0 * IMG;
            float* arow = &sA[p * A_STRIDE + kidx];
            #pragma unroll
            for (int cl = 0; cl < CI_CHUNK; ++cl) {
                const float g00 = v00 ? xc[o00] : 0.0f;
                const float g01 = v01 ? xc[o01] : 0.0f;
                const float g10 = v10 ? xc[o10] : 0.0f;
                const float g11 = v11 ? xc[o11] : 0.0f;
                arow[cl * 9] = g00 * w00 + g01 * w01 + g10 * w10 + g11 * w11;
                xc += IMG;
            }
        }
        __syncthreads();

        // Wave-level WMMA over the 72-deep K chunk.
        // A frag (16x4 f32): VGPR0 = K=k0 (lanes 0-15) / K=k0+2 (lanes 16-31),
        //                    VGPR1 = K=k0+1 / k0+3. M = lane&15.
        // B frag (4x16 f32): mirrored, N = lane&15. One aligned b64 LDS load per frag.
        #pragma unroll
        for (int k0 = 0; k0 < KC; k0 += 4) {
            const int kk = k0 + 2 * half;
            const int kp = (k0 >> 1) + half;          // K-pair row, == kk>>1
            const v2f a = { Arow[kk], Arow[kk + 1] };
            #pragma unroll
            for (int t = 0; t < 4; ++t) {
                const v2f b = *(const v2f*)&sB[kp * B_PS + ((t * 16 + row) << 1)];
                acc[t] = __builtin_amdgcn_wmma_f32_16x16x4_f32(
                    /*neg_a=*/false, a, /*neg_b=*/false, b,
                    /*c_mod=*/(short)0, acc[t],
                    /*reuse_a=*/false, /*reuse_b=*/false);
            }
        }
        __syncthreads();  // protect sA/sB before next chunk's fills
    }

    // ---------------- Phase 4: transpose through LDS, coalesced store ------------
    // C/D layout: VGPR r -> M = r + 8*half, N = lane&15.
    #pragma unroll
    for (int t = 0; t < 4; ++t) {
        const int co = t * 16 + row;
        #pragma unroll
        for (int r = 0; r < 8; ++r) {
            const int pl = wv * 16 + half * 8 + r;   // local pixel 0..63
            sA[co * 65 + pl] = acc[t][r];
        }
    }
    __syncthreads();
    for (int idx = tid; idx < C_OUT * NPIX_BLK; idx += THREADS) {
        const int co = idx >> 6;
        const int pl = idx & (NPIX_BLK - 1);
        out[((size_t)(n * C_OUT + co)) * IMG + pin0 + pl] = sA[co * 65 + pl] + b_conv[co];
    }
}

extern "C" void kernel_launch(void* const* d_in, const int* in_sizes, int n_in,
                              void* d_out, int out_size, void* d_ws, size_t ws_size,
                              hipStream_t stream) {
    (void)n_in; (void)out_size; (void)d_ws; (void)ws_size;
    const float* x      = (const float*)d_in[0];
    const float* w_off  = (const float*)d_in[1];
    const float* b_off  = (const float*)d_in[2];
    const float* w_conv = (const float*)d_in[3];
    const float* b_conv = (const float*)d_in[4];
    float* out = (float*)d_out;

    const int npix    = in_sizes[0] / C_IN;       // n * 128 * 128
    const int nblocks = npix / NPIX_BLK;          // 2048 for batch 8
    deform_conv_fused<<<nblocks, THREADS, 0, stream>>>(x, w_off, b_off,
                                                       w_conv, b_conv, out);
}